// FourierConv_77713138254082
// MI455X (gfx1250) — compile-verified
//
#include <hip/hip_runtime.h>
#include <math.h>

typedef float v2f __attribute__((ext_vector_type(2)));
typedef float v8f __attribute__((ext_vector_type(8)));

namespace {
constexpr int   kP     = 511;                 // padded transform length (2*256-1)
constexpr float kTwoPi = 6.283185307179586f;
}

// ---------------------------------------------------------------------------
// Generic DFT-as-GEMM pass:
//   C[m, n] = scale * Sum_k cis(sgn * 2*pi * ((m + mOff) * k mod 511)/511) * B[k, n]
// One wave computes TWO 16(M) x 16(N) f32 tiles (rows m..m+15 and m+16..m+31)
// via V_WMMA_F32_16X16X4_F32, marching K in steps of 4. Each B fragment load
// feeds up to 8 WMMAs. Twiddles are generated in-register with a complex
// rotation recurrence (zero twiddle memory traffic).
//
// Complex tensors are stored INTERLEAVED (float2 = {re, im}), so each lane
// fetches one complex element with a single global_load_b64 and stores with a
// single global_store_b64 (halves VMEM instruction count vs planar r/i).
//
// K4 = K/4 exact: callers pad K to a multiple of 4 (zero rows in B), so the
// hot loop has NO guards, NO divergence.
//
// N is addressed as n = (n2*c1 + n1)*8 + n0 with independent strides for B
// and C (element units), letting every pass use natural layouts, no transposes.
// ---------------------------------------------------------------------------
template<bool B_CPLX, bool C_CPLX, bool BIAS>
__global__ __launch_bounds__(256) void dft_gemm_kernel(
    const float* __restrict__ Br,  const float2* __restrict__ Bc,
    float* __restrict__ Cr,        float2* __restrict__ Cc,
    const float* __restrict__ bias,
    int K4, int mhalf, int mOff, float sgn, float scale,
    long bsk, long bs1, long bs2, int c1,
    long csm, long cs1, long cs2)
{
    const int lane = threadIdx.x & 31;
    const int wave = threadIdx.x >> 5;
    const int tile = blockIdx.x * 8 + wave;   // 8 waves per 256-thread block
    const int mtH = tile % mhalf;             // pair-of-M-tiles index
    const int nt  = tile / mhalf;

    // Column this lane covers in the B (4x16) and C (16x16) fragments.
    const int n  = nt * 16 + (lane & 15);
    const int n0 = n & 7;
    const int nq = n >> 3;
    const int n1 = nq % c1;
    const int n2 = nq / c1;
    const long colB = (long)n2 * bs2 + (long)n1 * bs1 + (long)n0;
    const long colC = (long)n2 * cs2 + (long)n1 * cs1 + (long)n0;

    // A fragment (16x4 f32): lanes 0-15 hold M rows for K={0,1}, lanes 16-31
    // the same M rows for K={2,3}.
    const int kp = (lane >> 4) * 2;

    v2f ar[2], ai[2];
    float w4r[2], w4i[2];
#pragma unroll
    for (int r = 0; r < 2; ++r) {
        const int m  = mtH * 32 + r * 16 + (lane & 15);
        const int mm = (m + mOff) % kP;
        float s, c;
        __sincosf(sgn * kTwoPi * (float)((mm * kp) % kP) * (1.0f / (float)kP), &s, &c);
        ar[r].x = c; ai[r].x = s;
        __sincosf(sgn * kTwoPi * (float)((mm * (kp + 1)) % kP) * (1.0f / (float)kP), &s, &c);
        ar[r].y = c; ai[r].y = s;
        __sincosf(sgn * kTwoPi * (float)((mm * 4) % kP) * (1.0f / (float)kP), &s, &c);
        w4r[r] = c; w4i[r] = s;               // per-step rotation cis(sgn*2pi*m*4/511)
    }

    v8f accP[2] = {{}, {}}, accQ[2] = {{}, {}}, accR[2] = {{}, {}}, accS[2] = {{}, {}};

    const float*  qr0 = nullptr; const float*  qr1 = nullptr;   // real-B path
    const float2* qc0 = nullptr; const float2* qc1 = nullptr;   // complex-B path
    if constexpr (B_CPLX) {
        qc0 = Bc + colB + (long)kp * bsk;
        qc1 = qc0 + bsk;
    } else {
        qr0 = Br + colB + (long)kp * bsk;
        qr1 = qr0 + bsk;
    }
    const long step = bsk << 2;

#pragma unroll 4
    for (int t = 0; t < K4; ++t) {
        v2f br, bi;
        if constexpr (B_CPLX) {
            const float2 L0 = qc0[0];          // single b64: (re, im) of k
            const float2 L1 = qc1[0];          // single b64: (re, im) of k+1
            qc0 += step; qc1 += step;
            br.x = L0.x; br.y = L1.x;
            bi.x = L0.y; bi.y = L1.y;
        } else {
            br.x = qr0[0]; br.y = qr1[0];
            qr0 += step; qr1 += step;
        }

        // Complex GEMM via real WMMAs: Cr = P - Q, Ci = R + S
#pragma unroll
        for (int r = 0; r < 2; ++r) {
            accP[r] = __builtin_amdgcn_wmma_f32_16x16x4_f32(false, ar[r], false, br, (short)0, accP[r], false, false);
            if constexpr (C_CPLX)
                accS[r] = __builtin_amdgcn_wmma_f32_16x16x4_f32(false, ai[r], false, br, (short)0, accS[r], false, false);
            if constexpr (B_CPLX)
                accQ[r] = __builtin_amdgcn_wmma_f32_16x16x4_f32(false, ai[r], false, bi, (short)0, accQ[r], false, false);
            if constexpr (B_CPLX && C_CPLX)
                accR[r] = __builtin_amdgcn_wmma_f32_16x16x4_f32(false, ar[r], false, bi, (short)0, accR[r], false, false);
        }

        // Advance twiddles by delta-k = 4: (ar + i*ai) *= (w4r + i*w4i)
#pragma unroll
        for (int r = 0; r < 2; ++r) {
            float tr = ar[r].x, ti = ai[r].x;
            ar[r].x = fmaf(tr, w4r[r], -ti * w4i[r]);
            ai[r].x = fmaf(tr, w4i[r],  ti * w4r[r]);
            tr = ar[r].y; ti = ai[r].y;
            ar[r].y = fmaf(tr, w4r[r], -ti * w4i[r]);
            ai[r].y = fmaf(tr, w4i[r],  ti * w4r[r]);
        }
    }

    // Store: C fragment VGPR j -> row base + j + 8*(lane>=16), col = n.
    const float bv = BIAS ? bias[n0] : 0.0f;
#pragma unroll
    for (int r = 0; r < 2; ++r) {
        const int rowOff = mtH * 32 + r * 16 + ((lane >> 4) << 3);
#pragma unroll
        for (int j = 0; j < 8; ++j) {
            const long addr = colC + (long)(rowOff + j) * csm;
            float vr = accP[r][j];
            if constexpr (B_CPLX) vr -= accQ[r][j];
            if constexpr (C_CPLX) {
                float vi = accS[r][j];
                if constexpr (B_CPLX) vi += accR[r][j];
                Cc[addr] = make_float2(scale * vr, scale * vi);   // single b64 store
            } else {
                Cr[addr] = fmaf(scale, vr, bv);
            }
        }
    }
}

// ---------------------------------------------------------------------------
// Per-frequency complex channel mixing (einsum 'bhwi,hwio->bhwo'), in place
// over the interleaved spectrum. One thread owns one (fh, fw, b) slot of 8
// channels. Spectrum layout: float2 [fh][(fw*8 + b)][c].
// Also zero-fills the padded fh==511 row so pass D can run K=512 unguarded.
// ---------------------------------------------------------------------------
__global__ __launch_bounds__(256) void freq_einsum_kernel(
    float2* __restrict__ xf, const float* __restrict__ kern, int total)
{
    const int tid = blockIdx.x * 256 + threadIdx.x;
    if (tid >= total) return;
    const int b  = tid & 7;
    const int t  = tid >> 3;
    const int fw = t % kP;
    const int fh = t / kP;                                  // 0..511 inclusive
    const long xbase = (long)fh * 32704 + (long)(fw * 8 + b) * 8;

    if (fh >= kP) {                                         // padded zero row
#pragma unroll
        for (int o = 0; o < 8; ++o) xf[xbase + o] = make_float2(0.0f, 0.0f);
        return;
    }

    const long kbase = ((long)fh * kP + fw) * 64;
    const float* kr = kern + kbase;                         // kernel[0, fh, fw, :, :]
    const float* ki = kern + (long)kP * kP * 64 + kbase;    // kernel[1, fh, fw, :, :]

    float xr[8], xi[8], yr[8], yi[8];
#pragma unroll
    for (int i = 0; i < 8; ++i) {
        const float2 v = xf[xbase + i];
        xr[i] = v.x; xi[i] = v.y;
    }
#pragma unroll
    for (int o = 0; o < 8; ++o) { yr[o] = 0.0f; yi[o] = 0.0f; }
#pragma unroll
    for (int i = 0; i < 8; ++i) {
#pragma unroll
        for (int o = 0; o < 8; ++o) {
            const float krv = kr[i * 8 + o];
            const float kiv = ki[i * 8 + o];
            yr[o] = fmaf(xr[i], krv, fmaf(-xi[i], kiv, yr[o]));
            yi[o] = fmaf(xr[i], kiv, fmaf( xi[i], krv, yi[o]));
        }
    }
#pragma unroll
    for (int o = 0; o < 8; ++o) xf[xbase + o] = make_float2(yr[o], yi[o]);
}

// Zero the padded fw==511 block of Y1 (float2 [256][512*64] layout) so pass E
// can run K=512 unguarded. 16384 threads.
__global__ __launch_bounds__(256) void y1_pad_zero_kernel(float2* __restrict__ y1)
{
    const int tid = blockIdx.x * 256 + threadIdx.x;
    const long addr = (long)(tid >> 6) * 32768 + 511L * 64 + (tid & 63);
    y1[addr] = make_float2(0.0f, 0.0f);
}

// ---------------------------------------------------------------------------
// Workspace map (float2 complex elements):
//   F1  : [512][16384]   (row-DFT of input; rows padded to 512)        67 MB
//   XF  : [512][32704]   (2-D spectrum; mixed in place; row 511 zero) 134 MB
//   Y1  : reuses F1 region ([256][32768], fw==511 block zeroed)        67 MB
// Total ~192 MiB; mostly L2-resident on MI455X (192 MB L2).
// ---------------------------------------------------------------------------
extern "C" void kernel_launch(void* const* d_in, const int* in_sizes, int n_in,
                              void* d_out, int out_size, void* d_ws, size_t ws_size,
                              hipStream_t stream) {
    (void)in_sizes; (void)n_in; (void)out_size; (void)ws_size;
    const float* x    = (const float*)d_in[0];   // (8,256,256,8)
    const float* kern = (const float*)d_in[1];   // (2,511,511,8,8)
    const float* bias = (const float*)d_in[2];   // (8,)
    float* out = (float*)d_out;                  // (8,256,256,8)

    float2* f1 = (float2*)d_ws;
    float2* xf = f1 + (long)512 * 16384;
    float2* y1 = f1;   // F1 dead after pass B -> reuse for Y1 (exact fit)

    // Pass A: row DFT along W. B = x real, addr = (b*256+h)*2048 + w*8 + cin.
    //   M=511(pad 512), K=256, N=16384 (b,h,cin). 16x1024 tile-pairs.
    dft_gemm_kernel<false, true, false><<<2048, 256, 0, stream>>>(
        x, nullptr, nullptr, f1, nullptr,
        64, 16, 0, -1.0f, 1.0f,
        8L, 2048L, 0L, 2048,
        16384L, 8L, 0L);

    // Pass B: column DFT along H. B = F1[fw][(b,h,cin)], k=h.
    //   M=511(pad 512), K=256, N=32704. 16x2044 tile-pairs.
    dft_gemm_kernel<true, true, false><<<4088, 256, 0, stream>>>(
        nullptr, f1, nullptr, xf, nullptr,
        64, 16, 0, -1.0f, 1.0f,
        8L, 2048L, 0L, 4088,
        32704L, 8L, 0L);

    // Pass C: per-frequency Cin->Cout complex mixing, in place over XF,
    //   plus zero-fill of the padded fh==511 row (grid covers fh in [0,512)).
    {
        const int total = 512 * kP * 8;   // 2,093,056 -> 8176 full blocks
        freq_einsum_kernel<<<total / 256, 256, 0, stream>>>(xf, kern, total);
    }

    // Zero padded fw==511 block of Y1 before pass E reads it.
    y1_pad_zero_kernel<<<64, 256, 0, stream>>>(y1);

    // Pass D: inverse column DFT with row crop (ho -> fh phase offset 127).
    //   M=256, K=512 (row 511 zero), N=32704. C row stride padded to 32768.
    dft_gemm_kernel<true, true, false><<<2044, 256, 0, stream>>>(
        nullptr, xf, nullptr, y1, nullptr,
        128, 8, 127, 1.0f, 1.0f,
        32704L, 8L, 0L, 4088,
        32768L, 8L, 0L);

    // Pass E: inverse row DFT with column crop, real part only, 1/511^2 scale,
    //   bias add, scattered store into (b,ho,wo,cout) output layout.
    //   M=256 (wo), K=512 (fw=511 block zero), N=16384 (ho,b,cout).
    dft_gemm_kernel<true, false, true><<<1024, 256, 0, stream>>>(
        nullptr, y1, out, nullptr, bias,
        128, 8, 127, 1.0f, 1.0f / ((float)kP * (float)kP),
        64L, 8L, 32768L, 8,
        8L, 524288L, 2048L);
}